// RelationGraphAttention_12189117186452
// MI455X (gfx1250) — compile-verified
//
#include <hip/hip_runtime.h>
#include <math.h>

typedef float v2f __attribute__((ext_vector_type(2)));
typedef float v8f __attribute__((ext_vector_type(8)));

constexpr int NREL = 3;
constexpr int BS   = 2;
constexpr int NN   = 2048;   // nodes
constexpr int CIN  = 256;
constexpr int CD   = 64;     // cout (head dim)
constexpr int NH   = 4;
constexpr int HO   = NH * CD;  // 256
constexpr float SLOPE = 0.1f;
constexpr float NEGINF = -1e18f;
constexpr int VS_STRIDE = 66;  // col-major V stage stride (even -> 8B-aligned b64 reads)

__device__ __forceinline__ v8f wmma_f32(v2f a, v2f b, v8f c) {
    // D = A(16x4,f32) * B(4x16,f32) + C(16x16,f32)
    return __builtin_amdgcn_wmma_f32_16x16x4_f32(
        false, a, false, b, (short)0, c, false, false);
}

// ---------------------------------------------------------------------------
// Generic fp32 WMMA GEMM: C[M,Nn] = A[M,K] @ B[K,Nn], row-major.
// One wave per 16x16 C tile, K-step 4 (V_WMMA_F32_16X16X4_F32).
// A-frag: lane<16 -> row m0+lane, k..k+1 ; lane>=16 -> row m0+lane-16, k+2..k+3
// B-frag: lane<16 -> col n0+lane, rows k,k+1 ; lane>=16 -> rows k+2,k+3
// ---------------------------------------------------------------------------
__global__ __launch_bounds__(128) void gemm16(const float* __restrict__ A,
                                              const float* __restrict__ B,
                                              float* __restrict__ C,
                                              int M, int K, int Nn) {
    const int lane = threadIdx.x & 31;
    const int wave = threadIdx.x >> 5;
    const int tile = blockIdx.x * 4 + wave;
    const int ntN  = Nn >> 4;
    const int tM   = tile / ntN;
    const int tN   = tile - tM * ntN;
    if (tM * 16 >= M) return;                 // wave-uniform
    const int m0 = tM << 4, n0 = tN << 4;
    const int lr = lane & 15, lk = lane >> 4;

    const float* arow = A + (size_t)(m0 + lr) * K + lk * 2;
    const float* bcol = B + n0 + lr + (size_t)(lk * 2) * Nn;

    v8f acc = {};
    for (int k = 0; k < K; k += 4) {
        v2f a = *(const v2f*)(arow + k);
        v2f b;
        b.x = bcol[(size_t)k * Nn];
        b.y = bcol[(size_t)k * Nn + Nn];
        acc = wmma_f32(a, b, acc);
    }
    float* crow = C + (size_t)(m0 + lk * 8) * Nn + n0 + lr;
#pragma unroll
    for (int t = 0; t < 8; ++t) crow[(size_t)t * Nn] = acc[t];
}

// ---------------------------------------------------------------------------
// ei[r,b,h,i] = hs[r,b,i,h,:] . al[r,:]   (and ej with ar)
// ---------------------------------------------------------------------------
__global__ void eiej_kernel(const float* __restrict__ hs,
                            const float* __restrict__ al,
                            const float* __restrict__ ar,
                            float* __restrict__ ei, float* __restrict__ ej) {
    int idx = blockIdx.x * blockDim.x + threadIdx.x;     // (r, bn, h)
    if (idx >= NREL * BS * NN * NH) return;
    int h  = idx & 3;
    int bn = (idx >> 2) & (BS * NN - 1);
    int r  = idx >> 14;                                  // /(4096*4)
    const float* row = hs + ((size_t)(r * BS * NN + bn)) * HO + h * CD;
    const float* alp = al + r * CD;
    const float* arp = ar + r * CD;
    float si = 0.f, sj = 0.f;
#pragma unroll 8
    for (int d = 0; d < CD; ++d) { si += row[d] * alp[d]; sj += row[d] * arp[d]; }
    int b = bn >> 11, i = bn & (NN - 1);
    int o = ((r * BS + b) * NH + h) * NN + i;
    ei[o] = si;
    ej[o] = sj;
}

// ---------------------------------------------------------------------------
// Pass A: per row i, per head h: softmax stats (max m, sum l) of the masked
// logits. One wave32 per (r,b,i); adjacency read once for all 4 heads.
// ---------------------------------------------------------------------------
__global__ __launch_bounds__(256) void stats_kernel(const float* __restrict__ adjs,
                                                    const float* __restrict__ ei,
                                                    const float* __restrict__ ej,
                                                    float2* __restrict__ ml) {
    const int wid  = blockIdx.x * 8 + (threadIdx.x >> 5);
    const int lane = threadIdx.x & 31;
    const int r  = wid >> 12;            // /4096
    const int bi = wid & 4095;
    const int b  = bi >> 11, i = bi & (NN - 1);
    const float* adjRow = adjs + ((size_t)((r * BS + b) * NN + i)) * NN;
    const int ehb = (r * BS + b) * NH * NN;

    float eiv[NH], m[NH], l[NH];
#pragma unroll
    for (int h = 0; h < NH; ++h) {
        eiv[h] = ei[ehb + h * NN + i];
        m[h] = -3.0e38f;
        l[h] = 0.f;
    }
    for (int j = lane; j < NN; j += 32) {
        float a = adjRow[j];
#pragma unroll
        for (int h = 0; h < NH; ++h) {
            float s0 = eiv[h] + ej[ehb + h * NN + j];
            s0 = (s0 > 0.f) ? s0 : SLOPE * s0;
            float s  = (a != 0.f) ? s0 : NEGINF;
            float mn = fmaxf(m[h], s);
            l[h] = l[h] * __expf(m[h] - mn) + __expf(s - mn);
            m[h] = mn;
        }
    }
    // wave32 reduction of (m,l) pairs
#pragma unroll
    for (int off = 16; off > 0; off >>= 1) {
#pragma unroll
        for (int h = 0; h < NH; ++h) {
            float om = __shfl_xor(m[h], off, 32);
            float ol = __shfl_xor(l[h], off, 32);
            float mn = fmaxf(m[h], om);
            l[h] = l[h] * __expf(m[h] - mn) + ol * __expf(om - mn);
            m[h] = mn;
        }
    }
    if (lane == 0) {
#pragma unroll
        for (int h = 0; h < NH; ++h)
            ml[ehb + h * NN + i] = make_float2(m[h], l[h]);
    }
}

// ---------------------------------------------------------------------------
// Pass B: fused P @ V with WMMA, V staged in LDS.
// Block = 8 waves, all sharing one (r,b,h); wave w owns 16-row i-tile.
// Per 64-j chunk: cooperative global->LDS stage of V (64 x 64, col-major,
// stride 66), then each wave builds P in the WMMA A-fragment layout in
// registers and multiplies against LDS B-fragments (ds_load_b64 per frag).
// ---------------------------------------------------------------------------
__global__ __launch_bounds__(256) void attn_kernel(const float* __restrict__ adjs,
                                                   const float* __restrict__ hs,
                                                   const float* __restrict__ ei,
                                                   const float* __restrict__ ej,
                                                   const float2* __restrict__ ml,
                                                   float* __restrict__ pout) {
    __shared__ float Vs[CD * VS_STRIDE];     // 64 cols x 64 rows, col-major

    const int tid  = threadIdx.x;
    const int lane = tid & 31;
    const int wid  = blockIdx.x * 8 + (tid >> 5);
    const int it   = wid & 127;          // 128 i-tiles of 16
    const int rest = wid >> 7;           // block-uniform (16 blocks per r,b,h)
    const int h = rest & 3;
    const int b = (rest >> 2) & 1;
    const int r = rest >> 3;
    const int i0 = it << 4;
    const int lr = lane & 15, lk = lane >> 4;
    const int iRow = i0 + lr;
    const int ehb = ((r * BS + b) * NH + h) * NN;

    const float  ei_i = ei[ehb + iRow];
    const float2 mlv  = ml[ehb + iRow];
    const float  mi   = mlv.x;
    const float  linv = 1.0f / mlv.y;    // l >= 1 always (max term contributes 1)

    const float* adjRow = adjs + ((size_t)((r * BS + b) * NN + iRow)) * NN;
    const float* ejp    = ej + ehb;
    const float* vblock = hs + ((size_t)(r * BS + b) * NN) * HO + h * CD;

    // cooperative stage mapping: 256 threads cover 64 rows x 64 cols
    const int srow = tid >> 2;           // 0..63 (row within chunk)
    const int scb  = (tid & 3) << 4;     // col base: 0,16,32,48

    v8f acc0 = {}, acc1 = {}, acc2 = {}, acc3 = {};
    for (int jc = 0; jc < NN; jc += 64) {
        __syncthreads();                 // previous chunk fully consumed
        const float* src = vblock + (size_t)(jc + srow) * HO + scb;
#pragma unroll
        for (int q = 0; q < 16; ++q)
            Vs[(scb + q) * VS_STRIDE + srow] = src[q];
        __syncthreads();

#pragma unroll 4
        for (int j = jc; j < jc + 64; j += 4) {
            const int jA = j + lk * 2;
            v2f av  = *(const v2f*)(adjRow + jA);   // 8B-aligned (jA even)
            v2f ejv = *(const v2f*)(ejp + jA);
            float s0 = ei_i + ejv.x; s0 = (s0 > 0.f) ? s0 : SLOPE * s0;
            float s1 = ei_i + ejv.y; s1 = (s1 > 0.f) ? s1 : SLOPE * s1;
            v2f a;
            a.x = (av.x != 0.f) ? __expf(s0 - mi) * linv : 0.f;
            a.y = (av.y != 0.f) ? __expf(s1 - mi) * linv : 0.f;

            const int jj = (j - jc) + lk * 2;       // rows jj, jj+1 of chunk
            const float* vs0 = &Vs[lr * VS_STRIDE + jj];
            v2f b0 = *(const v2f*)(vs0);
            v2f b1 = *(const v2f*)(vs0 + 16 * VS_STRIDE);
            v2f b2 = *(const v2f*)(vs0 + 32 * VS_STRIDE);
            v2f b3 = *(const v2f*)(vs0 + 48 * VS_STRIDE);
            acc0 = wmma_f32(a, b0, acc0);
            acc1 = wmma_f32(a, b1, acc1);
            acc2 = wmma_f32(a, b2, acc2);
            acc3 = wmma_f32(a, b3, acc3);
        }
    }
    // C/D layout: VGPR t, lane -> row i0 + lk*8 + t, col d0 + lr
    float* orow = pout + ((size_t)(ehb + i0 + lk * 8)) * CD + lr;
#pragma unroll
    for (int t = 0; t < 8; ++t) {
        orow[(size_t)t * CD + 0]  = acc0[t];
        orow[(size_t)t * CD + 16] = acc1[t];
        orow[(size_t)t * CD + 32] = acc2[t];
        orow[(size_t)t * CD + 48] = acc3[t];
    }
}

// ---------------------------------------------------------------------------
// Sum over relations + residual projection, write (b, n, nh*d) output.
// ---------------------------------------------------------------------------
__global__ void finalize_kernel(const float* __restrict__ pout,
                                const float* __restrict__ xout,
                                float* __restrict__ out) {
    int idx = blockIdx.x * blockDim.x + threadIdx.x;  // b<<19 | i<<8 | h<<6 | d
    if (idx >= BS * NN * HO) return;
    int dd = idx & 63;
    int h  = (idx >> 6) & 3;
    int i  = (idx >> 8) & (NN - 1);
    int b  = idx >> 19;
    float s = xout[(size_t)(b * NN + i) * CD + dd];
#pragma unroll
    for (int r = 0; r < NREL; ++r)
        s += pout[((size_t)(((r * BS + b) * NH + h) * NN + i)) * CD + dd];
    out[idx] = s;
}

extern "C" void kernel_launch(void* const* d_in, const int* in_sizes, int n_in,
                              void* d_out, int out_size, void* d_ws, size_t ws_size,
                              hipStream_t stream) {
    const float* xs   = (const float*)d_in[0];   // (2,2048,256)
    const float* adjs = (const float*)d_in[1];   // (3,2,2048,2048)
    const float* W    = (const float*)d_in[2];   // (3,256,256)
    const float* al   = (const float*)d_in[3];   // (3,64)
    const float* ar   = (const float*)d_in[4];   // (3,64)
    const float* Wout = (const float*)d_in[5];   // (256,64)
    float* out        = (float*)d_out;

    float* ws   = (float*)d_ws;
    float* hs   = ws;                                   // 3*4096*256
    float* xout = hs + (size_t)NREL * BS * NN * HO;     // 4096*64
    float* ei   = xout + (size_t)BS * NN * CD;          // 3*2*4*2048
    float* ej   = ei + (size_t)NREL * BS * NH * NN;
    float2* ml  = (float2*)(ej + (size_t)NREL * BS * NH * NN);
    float* pout = (float*)(ml + (size_t)NREL * BS * NH * NN);  // 3*2*4*2048*64

    // hs[r] = X @ W[r]  (fp32 WMMA GEMMs)
    for (int r = 0; r < NREL; ++r) {
        int tiles = (BS * NN / 16) * (HO / 16);          // 4096
        gemm16<<<tiles / 4, 128, 0, stream>>>(xs, W + (size_t)r * CIN * HO,
                                              hs + (size_t)r * BS * NN * HO,
                                              BS * NN, CIN, HO);
    }
    {   // xout = X @ W_out
        int tiles = (BS * NN / 16) * (CD / 16);          // 1024
        gemm16<<<tiles / 4, 128, 0, stream>>>(xs, Wout, xout, BS * NN, CIN, CD);
    }
    eiej_kernel<<<(NREL * BS * NN * NH + 255) / 256, 256, 0, stream>>>(hs, al, ar, ei, ej);
    stats_kernel<<<NREL * BS * NN / 8, 256, 0, stream>>>(adjs, ei, ej, ml);
    attn_kernel<<<NREL * BS * NH * (NN / 16) / 8, 256, 0, stream>>>(adjs, hs, ei, ej, ml, pout);
    finalize_kernel<<<(BS * NN * HO + 255) / 256, 256, 0, stream>>>(pout, xout, out);
}